// GCNRefiner_42906723287163
// MI455X (gfx1250) — compile-verified
//
#include <hip/hip_runtime.h>
#include <hip/hip_bf16.h>

#define DIM 512

typedef float v2f __attribute__((ext_vector_type(2)));
typedef float v8f __attribute__((ext_vector_type(8)));

// ---------------------------------------------------------------------------
// Degree / normalization prep
// ---------------------------------------------------------------------------
__global__ void deg_init_kernel(float* deg, int n) {
    int i = blockIdx.x * blockDim.x + threadIdx.x;
    if (i < n) deg[i] = 1.0f;               // self-loop contributes 1
}

__global__ void deg_accum_kernel(const int* __restrict__ dst, float* deg, int E) {
    int i = blockIdx.x * blockDim.x + threadIdx.x;
    if (i < E) unsafeAtomicAdd(&deg[dst[i]], 1.0f);
}

__global__ void dinv_kernel(const float* __restrict__ deg, float* dinv, int n) {
    int i = blockIdx.x * blockDim.x + threadIdx.x;
    if (i < n) {
        float d = deg[i];
        dinv[i] = (d > 0.0f) ? rsqrtf(d) : 0.0f;  // deg >= 1 always (self loop)
    }
}

// ---------------------------------------------------------------------------
// FP32 WMMA GEMM:  C[M,512] = A[M,512] @ B[512,512]   (row major)
// One wave -> 16(M) x 64(N) tile, 4x v_wmma_f32_16x16x4_f32 accumulators.
// block = 128 threads (4 waves, each a different 64-wide N slice)
// grid  = (M/16, 512/256)
// ---------------------------------------------------------------------------
__global__ __launch_bounds__(128) void gemm_f32_wmma_kernel(
        const float* __restrict__ A, const float* __restrict__ B,
        float* __restrict__ C, int M) {
    const int lane  = threadIdx.x & 31;
    const int wave  = threadIdx.x >> 5;
    const int half  = lane >> 4;     // 0 => lanes 0-15, 1 => lanes 16-31
    const int l16   = lane & 15;
    const int mBase = blockIdx.x * 16;
    const int nBase = blockIdx.y * 256 + wave * 64;
    if (mBase >= M) return;

    v8f acc0 = {}, acc1 = {}, acc2 = {}, acc3 = {};

    // A fragment (16x4 f32): lane<16 holds {K,K+1} of row M=l16, lane>=16 holds {K+2,K+3}
    const float* Arow = A + (size_t)(mBase + l16) * DIM;

    for (int k = 0; k < DIM; k += 4) {
        const int ka = k + half * 2;
        v2f a;
        a.x = Arow[ka + 0];
        a.y = Arow[ka + 1];

        // B fragment (4x16 f32), mirrored layout: VGPR0 rows {k, k+2}, VGPR1 rows {k+1, k+3}
        const float* Bb = B + (size_t)ka * DIM + nBase + l16;
        v2f b0, b1, b2, b3;
        b0.x = Bb[0];        b0.y = Bb[DIM];
        b1.x = Bb[16];       b1.y = Bb[DIM + 16];
        b2.x = Bb[32];       b2.y = Bb[DIM + 32];
        b3.x = Bb[48];       b3.y = Bb[DIM + 48];

        acc0 = __builtin_amdgcn_wmma_f32_16x16x4_f32(false, a, false, b0, (short)0, acc0, false, false);
        acc1 = __builtin_amdgcn_wmma_f32_16x16x4_f32(false, a, false, b1, (short)0, acc1, false, false);
        acc2 = __builtin_amdgcn_wmma_f32_16x16x4_f32(false, a, false, b2, (short)0, acc2, false, false);
        acc3 = __builtin_amdgcn_wmma_f32_16x16x4_f32(false, a, false, b3, (short)0, acc3, false, false);
    }

    // C/D layout: VGPR j, lanes 0-15 -> row mBase+j, lanes 16-31 -> row mBase+8+j; col = nBase+l16
    float* Crow = C + (size_t)(mBase + half * 8) * DIM + nBase + l16;
#pragma unroll
    for (int j = 0; j < 8; ++j) {
        Crow[(size_t)j * DIM + 0]  = acc0[j];
        Crow[(size_t)j * DIM + 16] = acc1[j];
        Crow[(size_t)j * DIM + 32] = acc2[j];
        Crow[(size_t)j * DIM + 48] = acc3[j];
    }
}

// ---------------------------------------------------------------------------
// agg[i, :] = bias[:]   (bias folded into accumulator init)
// ---------------------------------------------------------------------------
__global__ void row_init_kernel(float* __restrict__ agg, const float* __restrict__ bias,
                                size_t total) {
    size_t i = (size_t)blockIdx.x * blockDim.x + threadIdx.x;
    if (i < total) agg[i] = bias[i & (DIM - 1)];
}

// ---------------------------------------------------------------------------
// Edge scatter: agg[dst] += h[src] * dinv[src] * dinv[dst]
// 64 threads per edge, 8 floats per thread; edges [E, E+N) are self loops.
// ---------------------------------------------------------------------------
__global__ __launch_bounds__(256) void scatter_kernel(
        const float* __restrict__ h, const int* __restrict__ src,
        const int* __restrict__ dst, const float* __restrict__ dinv,
        float* __restrict__ agg, int E, int total) {
    int edge = blockIdx.x * 4 + (threadIdx.x >> 6);
    if (edge >= total) return;
    int t = threadIdx.x & 63;

    int s, d;
    if (edge < E) { s = src[edge]; d = dst[edge]; }
    else          { s = d = edge - E; }            // self loop

    float norm = dinv[s] * dinv[d];

    const float4* hp = (const float4*)(h + (size_t)s * DIM) + t * 2;
    float*        ap = agg + (size_t)d * DIM + t * 8;
    float4 v0 = hp[0];
    float4 v1 = hp[1];

    unsafeAtomicAdd(ap + 0, v0.x * norm);
    unsafeAtomicAdd(ap + 1, v0.y * norm);
    unsafeAtomicAdd(ap + 2, v0.z * norm);
    unsafeAtomicAdd(ap + 3, v0.w * norm);
    unsafeAtomicAdd(ap + 4, v1.x * norm);
    unsafeAtomicAdd(ap + 5, v1.y * norm);
    unsafeAtomicAdd(ap + 6, v1.z * norm);
    unsafeAtomicAdd(ap + 7, v1.w * norm);
}

__global__ void relu_kernel(float* __restrict__ v, size_t total) {
    size_t i = (size_t)blockIdx.x * blockDim.x + threadIdx.x;
    if (i < total) v[i] = fmaxf(v[i], 0.0f);
}

// ---------------------------------------------------------------------------
// Final: y = x + clip(alpha)*h2 ; out = y / max(||y||, 1e-12)
// One block (128 threads, 4 waves) per node; float4 per thread.
// ---------------------------------------------------------------------------
__global__ __launch_bounds__(128) void final_kernel(
        const float* __restrict__ x, const float* __restrict__ agg,
        const float* __restrict__ alpha_p, float* __restrict__ out) {
    const int node = blockIdx.x;
    const int t = threadIdx.x;

    float a = alpha_p[0];
    a = fminf(1.0f, fmaxf(-1.0f, a));

    const float4* xp = (const float4*)(x   + (size_t)node * DIM);
    const float4* gp = (const float4*)(agg + (size_t)node * DIM);
    float4 xv = xp[t];
    float4 gv = gp[t];

    float4 y;
    y.x = xv.x + a * gv.x;
    y.y = xv.y + a * gv.y;
    y.z = xv.z + a * gv.z;
    y.w = xv.w + a * gv.w;

    float ss = y.x * y.x + y.y * y.y + y.z * y.z + y.w * y.w;
#pragma unroll
    for (int off = 16; off > 0; off >>= 1)
        ss += __shfl_down(ss, off, 32);

    __shared__ float red[4];
    if ((threadIdx.x & 31) == 0) red[threadIdx.x >> 5] = ss;
    __syncthreads();
    float tot = red[0] + red[1] + red[2] + red[3];

    float inv = 1.0f / fmaxf(sqrtf(tot), 1e-12f);
    float4 o;
    o.x = y.x * inv; o.y = y.y * inv; o.z = y.z * inv; o.w = y.w * inv;
    ((float4*)out)[(size_t)node * (DIM / 4) + t] = o;
}

// ---------------------------------------------------------------------------
// Host launch
// ---------------------------------------------------------------------------
extern "C" void kernel_launch(void* const* d_in, const int* in_sizes, int n_in,
                              void* d_out, int out_size, void* d_ws, size_t ws_size,
                              hipStream_t stream) {
    const float* x     = (const float*)d_in[0];
    const int*   ei    = (const int*)d_in[1];
    const float* W1    = (const float*)d_in[2];
    const float* b1    = (const float*)d_in[3];
    const float* W2    = (const float*)d_in[4];
    const float* b2    = (const float*)d_in[5];
    const float* alpha = (const float*)d_in[6];

    const int N = in_sizes[0] / DIM;      // 20000
    const int E = in_sizes[1] / 2;        // 320000
    const int total_edges = E + N;        // with self loops
    const size_t ND = (size_t)N * DIM;

    // workspace layout
    float* deg  = (float*)d_ws;
    float* dinv = deg + N;
    float* hbuf = (float*)(((uintptr_t)(dinv + N) + 255) & ~(uintptr_t)255);
    float* agg  = hbuf + ND;

    const int* src = ei;
    const int* dst = ei + E;

    // 1) symmetric normalization coefficients
    deg_init_kernel<<<(N + 255) / 256, 256, 0, stream>>>(deg, N);
    deg_accum_kernel<<<(E + 255) / 256, 256, 0, stream>>>(dst, deg, E);
    dinv_kernel<<<(N + 255) / 256, 256, 0, stream>>>(deg, dinv, N);

    dim3 gemm_grid(N / 16, DIM / 256);
    dim3 gemm_block(128);
    int scat_grid = (total_edges + 3) / 4;
    int elem_grid = (int)((ND + 255) / 256);

    // 2) conv1: h = x@W1 ; agg = b1 + scatter ; relu
    gemm_f32_wmma_kernel<<<gemm_grid, gemm_block, 0, stream>>>(x, W1, hbuf, N);
    row_init_kernel<<<elem_grid, 256, 0, stream>>>(agg, b1, ND);
    scatter_kernel<<<scat_grid, 256, 0, stream>>>(hbuf, src, dst, dinv, agg, E, total_edges);
    relu_kernel<<<elem_grid, 256, 0, stream>>>(agg, ND);

    // 3) conv2: h = relu(agg)@W2 ; agg = b2 + scatter
    gemm_f32_wmma_kernel<<<gemm_grid, gemm_block, 0, stream>>>(agg, W2, hbuf, N);
    row_init_kernel<<<elem_grid, 256, 0, stream>>>(agg, b2, ND);
    scatter_kernel<<<scat_grid, 256, 0, stream>>>(hbuf, src, dst, dinv, agg, E, total_edges);

    // 4) residual + L2 normalize
    final_kernel<<<N, 128, 0, stream>>>(x, agg, alpha, (float*)d_out);
}